// ConformerAggrEncoderV2_41875931136505
// MI455X (gfx1250) — compile-verified
//
#include <hip/hip_runtime.h>
#include <hip/hip_bf16.h>
#include <math.h>

#define BB 32
#define TT 1024
#define FF 512
#define EPSV 1e-6f

typedef __attribute__((ext_vector_type(2))) float v2f;
typedef __attribute__((ext_vector_type(8))) float v8f;

// ---------------------------------------------------------------------------
// Kernel 1: gate GEMV via V_WMMA_F32_16X16X4_F32.
// One wave handles a 16-row tile of x (16 time steps), accumulating over
// K=512 in steps of 4. B operand = aggr_w chunk replicated across all 16
// columns, so D[m][n] == dot(x_row_m, w) for every n.
// A layout (32-bit 16x4): lane = 16*g + m holds K = 2g (vgpr0), 2g+1 (vgpr1).
// D layout (32-bit 16x16): vgpr r, lanes 0-15 -> M=r, lanes 16-31 -> M=8+r.
// ---------------------------------------------------------------------------
__global__ __launch_bounds__(128) void gate_wmma_kernel(
    const float* __restrict__ x, const float* __restrict__ w,
    const float* __restrict__ bias, float* __restrict__ ws_gate) {
  const int lane = threadIdx.x & 31;
  const int wave = threadIdx.x >> 5;
  const int tile = blockIdx.x * 4 + wave;          // (B*T)/16 tiles total
  const int r0 = tile * 16;
  const int m = lane & 15;
  const int g = lane >> 4;                          // 0 or 1
  const float* __restrict__ xrow = x + (size_t)(r0 + m) * FF;

  v8f acc = {};
#pragma unroll 4
  for (int k0 = 0; k0 < FF; k0 += 4) {
    v2f a, bv;
    a[0]  = xrow[k0 + 2 * g];
    a[1]  = xrow[k0 + 2 * g + 1];
    bv[0] = w[k0 + 2 * g];
    bv[1] = w[k0 + 2 * g + 1];
    acc = __builtin_amdgcn_wmma_f32_16x16x4_f32(
        /*neg_a=*/false, a, /*neg_b=*/false, bv,
        /*c_mod=*/(short)0, acc, /*reuse_a=*/false, /*reuse_b=*/false);
  }

  const float bb = bias[0];
  if (m == 0) {   // lane 0 holds M=0..7, lane 16 holds M=8..15
#pragma unroll
    for (int r = 0; r < 8; ++r) {
      float z = acc[r] + bb;
      ws_gate[r0 + g * 8 + r] = 1.0f / (1.0f + __expf(-z));
    }
  }
}

// ---------------------------------------------------------------------------
// Block-wide inclusive scans (blockDim.x == TT, called uniformly)
// ---------------------------------------------------------------------------
__device__ inline void scan_inc_f(float* s, int t) {
  for (int off = 1; off < TT; off <<= 1) {
    float add = (t >= off) ? s[t - off] : 0.0f;
    __syncthreads();
    s[t] += add;
    __syncthreads();
  }
}
__device__ inline void scan_inc_i(int* s, int t) {
  for (int off = 1; off < TT; off <<= 1) {
    int add = (t >= off) ? s[t - off] : 0;
    __syncthreads();
    s[t] += add;
    __syncthreads();
  }
}

// ---------------------------------------------------------------------------
// Kernel 2: per-batch valley detection, segment compaction, prefix-sum
// denominators. One block (TT threads) per batch.
// ---------------------------------------------------------------------------
__global__ __launch_bounds__(TT) void segment_detect_kernel(
    const float* __restrict__ ws_gate, int* __restrict__ seg_start,
    int* __restrict__ seg_end, float* __restrict__ den,
    int* __restrict__ counts) {
  __shared__ float wsh[TT];
  __shared__ float fscan[TT];
  __shared__ int   iscan[TT];
  __shared__ int   sstart[TT];
  __shared__ int   send[TT];
  __shared__ int   totals;

  const int b = blockIdx.x;
  const int t = threadIdx.x;

  wsh[t] = ws_gate[b * TT + t];
  __syncthreads();

  const float wd = wsh[t];
  const float sb = (t > 0)      ? wsh[t - 1] : 0.0f;
  const float sa = (t < TT - 1) ? wsh[t + 1] : 0.0f;
  const bool valley = (wd < sb) && (wd < sa);
  const bool startm = valley || (t == 0);
  const bool endm   = (valley && (t != 0)) || (t == TT - 1);

  // compaction rank for starts
  iscan[t] = startm ? 1 : 0;
  __syncthreads();
  scan_inc_i(iscan, t);
  if (startm) sstart[iscan[t] - 1] = t;
  if (t == TT - 1) totals = iscan[t];
  __syncthreads();

  // compaction rank for ends
  iscan[t] = endm ? 1 : 0;
  __syncthreads();
  scan_inc_i(iscan, t);
  if (endm) send[iscan[t] - 1] = min(t + 2, TT);
  __syncthreads();

  // inclusive prefix of ws -> segment denominators
  fscan[t] = wd;
  __syncthreads();
  scan_inc_f(fscan, t);

  const int total = totals;
  if (t < total) {
    const int st = sstart[t];
    const int en = send[t];
    const float d = fscan[en - 1] - ((st > 0) ? fscan[st - 1] : 0.0f);
    seg_start[b * TT + t] = st;
    seg_end[b * TT + t]   = en;
    den[b * TT + t]       = fmaxf(d, EPSV);
  } else {
    seg_start[b * TT + t] = 0;
    seg_end[b * TT + t]   = 0;
    den[b * TT + t]       = EPSV;
  }
  if (t == 0) counts[b] = total;
}

// ---------------------------------------------------------------------------
// Kernel 3: max(counts) across batches, new_mask = s < lengths/lengths0*maxc
// ---------------------------------------------------------------------------
__global__ __launch_bounds__(256) void finalize_mask_kernel(
    const int* __restrict__ counts, const int* __restrict__ seq_len,
    float* __restrict__ new_mask) {
  __shared__ int red[256];
  const int t = threadIdx.x;
  red[t] = (t < BB) ? counts[t] : 0;
  __syncthreads();
  for (int off = 128; off > 0; off >>= 1) {
    if (t < off) red[t] = max(red[t], red[t + off]);
    __syncthreads();
  }
  const float maxc = (float)red[0];
  const float len0 = (float)seq_len[0];
  for (int i = t; i < BB * TT; i += 256) {
    const int b = i / TT;
    const int s = i - b * TT;
    const int nl = (int)((float)seq_len[b] / len0 * maxc);
    new_mask[i] = (s < nl) ? 1.0f : 0.0f;
  }
}

// ---------------------------------------------------------------------------
// Kernel 4: segment-weighted means. One block per (segment, batch).
// 128 threads x float4 = 512 features. Total x traffic ~1.2x (segments
// overlap by <=2 rows). Invalid segment rows are zero-filled.
// ---------------------------------------------------------------------------
__global__ __launch_bounds__(128) void segment_mean_kernel(
    const float* __restrict__ x, const float* __restrict__ ws_gate,
    const int* __restrict__ seg_start, const int* __restrict__ seg_end,
    const float* __restrict__ den, const int* __restrict__ counts,
    float* __restrict__ out) {
  const int s = blockIdx.x;
  const int b = blockIdx.y;
  const int f4 = threadIdx.x;  // owns features 4*f4 .. 4*f4+3

  float4 acc = make_float4(0.f, 0.f, 0.f, 0.f);
  if (s < counts[b]) {
    const int st = seg_start[b * TT + s];
    const int en = seg_end[b * TT + s];
    for (int t = st; t < en; ++t) {
      const float w = ws_gate[b * TT + t];
      const float4 xv =
          ((const float4*)(x + ((size_t)(b * TT + t)) * FF))[f4];
      acc.x += w * xv.x;
      acc.y += w * xv.y;
      acc.z += w * xv.z;
      acc.w += w * xv.w;
    }
    const float inv = 1.0f / den[b * TT + s];
    acc.x *= inv; acc.y *= inv; acc.z *= inv; acc.w *= inv;
  }
  ((float4*)(out + ((size_t)(b * TT + s)) * FF))[f4] = acc;
}

// ---------------------------------------------------------------------------
extern "C" void kernel_launch(void* const* d_in, const int* in_sizes, int n_in,
                              void* d_out, int out_size, void* d_ws,
                              size_t ws_size, hipStream_t stream) {
  const float* x      = (const float*)d_in[0];   // [B,T,F]
  const float* aggr_w = (const float*)d_in[1];   // [F,1]
  const float* aggr_b = (const float*)d_in[2];   // [1]
  const int*   seqlen = (const int*)d_in[3];     // [B]

  float* out      = (float*)d_out;                       // [B,T,F]
  float* new_mask = (float*)d_out + (size_t)BB * TT * FF; // [B,T]

  // workspace layout
  char* ws = (char*)d_ws;
  float* ws_gate   = (float*)(ws + 0);                         // B*T f32
  int*   seg_start = (int*)(ws + (size_t)BB * TT * 4);         // B*T i32
  int*   seg_end   = (int*)(ws + (size_t)BB * TT * 8);         // B*T i32
  float* den       = (float*)(ws + (size_t)BB * TT * 12);      // B*T f32
  int*   counts    = (int*)(ws + (size_t)BB * TT * 16);        // B   i32

  // K1: gate = sigmoid(x @ w + b), WMMA f32 16x16x4, 4 tiles/block
  const int n_tiles = (BB * TT) / 16;  // 2048
  gate_wmma_kernel<<<n_tiles / 4, 128, 0, stream>>>(x, aggr_w, aggr_b,
                                                    ws_gate);

  // K2: valleys -> compacted segments + denominators
  segment_detect_kernel<<<BB, TT, 0, stream>>>(ws_gate, seg_start, seg_end,
                                               den, counts);

  // K3: max counts + new sequence mask
  finalize_mask_kernel<<<1, 256, 0, stream>>>(counts, seqlen, new_mask);

  // K4: segment-weighted means
  dim3 grid4(TT, BB);
  segment_mean_kernel<<<grid4, 128, 0, stream>>>(x, ws_gate, seg_start,
                                                 seg_end, den, counts, out);
}